// RandomRectangleErasing_40870908789026
// MI455X (gfx1250) — compile-verified
//
#include <hip/hip_runtime.h>

// Random rectangle erasing: out = img * (in_rect ? 0 : 1)
// B=64, C=3, H=512, W=512, fp32. Pure streaming kernel:
// 402.7 MB total traffic -> ~17.3 us floor at 23.3 TB/s HBM.

typedef float v4f __attribute__((ext_vector_type(4)));

constexpr int Bsz = 64;
constexpr int Cch = 3;
constexpr int Hh  = 512;
constexpr int Ww  = 512;
constexpr int ROWS = 4;              // rows of one (b) handled per block
constexpr int W4  = Ww / 4;          // float4 per row = 128 = blockDim.x

__global__ __launch_bounds__(128)
void erase_rect_kernel(const float* __restrict__ img,
                       const int*   __restrict__ widths,
                       const int*   __restrict__ heights,
                       const int*   __restrict__ xs,
                       const int*   __restrict__ ys,
                       float*       __restrict__ out)
{
    const int b   = blockIdx.z;              // uniform -> scalar regs
    const int ch0 = blockIdx.y * ROWS;       // c*H + h base, uniform
    const int w0  = threadIdx.x << 2;        // first of 4 pixels this lane owns

    // Per-batch rect params: uniform index -> compiler emits s_load_b32.
    const int x0 = xs[b];
    const int y0 = ys[b];
    const int x1 = x0 + widths[b];
    const int y1 = y0 + heights[b];

    const long long base = ((long long)b * (Cch * Hh) + ch0) * (long long)Ww + w0;
    const v4f* __restrict__ src = (const v4f*)(img + base);
    v4f*       __restrict__ dst = (v4f*)(out + base);

    // Issue all 4 row loads first (independent b128 NT loads -> MLP).
    v4f v[ROWS];
#pragma unroll
    for (int r = 0; r < ROWS; ++r)
        v[r] = __builtin_nontemporal_load(src + r * W4);

#pragma unroll
    for (int r = 0; r < ROWS; ++r) {
        const int h = (ch0 + r) & (Hh - 1);  // row within image (c irrelevant to mask)
        // Scalar (wave-uniform) branch: most rows skip the per-lane masking entirely.
        if (h >= y0 && h < y1) {
            if (w0 + 0 >= x0 && w0 + 0 < x1) v[r].x = 0.0f;
            if (w0 + 1 >= x0 && w0 + 1 < x1) v[r].y = 0.0f;
            if (w0 + 2 >= x0 && w0 + 2 < x1) v[r].z = 0.0f;
            if (w0 + 3 >= x0 && w0 + 3 < x1) v[r].w = 0.0f;
        }
        __builtin_nontemporal_store(v[r], dst + r * W4);
    }
}

extern "C" void kernel_launch(void* const* d_in, const int* in_sizes, int n_in,
                              void* d_out, int out_size, void* d_ws, size_t ws_size,
                              hipStream_t stream)
{
    (void)in_sizes; (void)n_in; (void)out_size; (void)d_ws; (void)ws_size;

    const float* img     = (const float*)d_in[0];
    const int*   widths  = (const int*)  d_in[1];
    const int*   heights = (const int*)  d_in[2];
    const int*   xs      = (const int*)  d_in[3];
    const int*   ys      = (const int*)  d_in[4];
    float*       out     = (float*)d_out;

    dim3 block(128, 1, 1);                        // 4 wave32
    dim3 grid(1, (Cch * Hh) / ROWS, Bsz);         // 384 x 64 = 24576 blocks, 8KB each

    erase_rect_kernel<<<grid, block, 0, stream>>>(img, widths, heights, xs, ys, out);
}